// Enhanced_SOTA_Temporal_PGAT_MoE_63393717289137
// MI455X (gfx1250) — compile-verified
//
#include <hip/hip_runtime.h>
#include <math.h>

// ---------------- problem constants (match reference) ----------------
#define BB   4
#define NN   256
#define DD   512
#define HH   8
#define HD   64
#define LL   3
#define KSP  25          // top-k edges per node
#define EPS_LN 1e-5f

typedef __attribute__((ext_vector_type(2))) float v2f;
typedef __attribute__((ext_vector_type(8))) float v8f;

// exact pointee type the async-LDS builtins expect: GCC-vector int4
typedef int v4i_vs __attribute__((vector_size(16)));
#define AS1 __attribute__((address_space(1)))
#define AS3 __attribute__((address_space(3)))

// -------- CDNA5 async global->LDS path (guarded; fallback = plain LDS) ------
#if defined(__gfx1250__) && __has_builtin(__builtin_amdgcn_global_load_async_to_lds_b128)
#define HAVE_ASYNC 1
#else
#define HAVE_ASYNC 0
#endif

template <int N>
__device__ __forceinline__ void async_wait()
{
#if defined(__gfx1250__)
#if __has_builtin(__builtin_amdgcn_s_wait_asynccnt)
    __builtin_amdgcn_s_wait_asynccnt(N);
#else
    asm volatile("s_wait_asynccnt %0" ::"i"(N) : "memory");
#endif
#endif
}

// =====================================================================
// WMMA fp32 GEMM, 32x32 macro-tile per 128-thread block (4 waves, 2x2
// grid of 16x16 WMMA tiles). K blocked by 32, A/B blocks staged in LDS
// via async global->LDS B128 copies, double-buffered so the next
// block's DMA overlaps the WMMA stream. LDS rows padded to 33 floats.
//   C[M,N] = A[M,K] @ W[K,N] (+bias) (+relu);  M,N,K multiples of 32.
// =====================================================================
#define BKM  32
#define BKK  32
#define LDSP 33

__global__ __launch_bounds__(128)
void gemm32_wmma(const float* __restrict__ A, int ldA,
                 const float* __restrict__ W, int Ncols, int K,
                 const float* __restrict__ bias,
                 float* __restrict__ C, int ldC,
                 int tiles32N, int relu)
{
    __shared__ float As[2][BKM * LDSP];
    __shared__ float Bs[2][BKK * LDSP];

    int tm0 = (blockIdx.x / tiles32N) * BKM;
    int tn0 = (blockIdx.x % tiles32N) * BKM;

    int t    = threadIdx.x;
    int lane = t & 31;
    int wv   = t >> 5;
    int wm   = wv >> 1, wn = wv & 1;          // 2x2 wave grid
    int half = lane >> 4, lr = lane & 15;

    int nkb = K / BKK;

    // Stage one 32x32 A block + one 32x32 B block: 128 threads x
    // (2 float4 A + 2 float4 B) => 4 async instructions per wave.
    auto issue = [&](int buf, int kb) {
        int k0 = kb * BKK;
#pragma unroll
        for (int e0 = 0; e0 < 2; ++e0) {
            int e   = t + e0 * 128;           // 0..255 float4 slots
            int row = e >> 3;                 // 32 rows
            int c4  = (e & 7) * 4;            // col within row
            const float* ga = A + (size_t)(tm0 + row) * ldA + k0 + c4;
            const float* gb = W + (size_t)(k0 + row) * Ncols + tn0 + c4;
            float* la = &As[buf][row * LDSP + c4];
            float* lb = &Bs[buf][row * LDSP + c4];
#if HAVE_ASYNC
            __builtin_amdgcn_global_load_async_to_lds_b128(
                (AS1 v4i_vs*)ga, (AS3 v4i_vs*)la, 0, 0);
            __builtin_amdgcn_global_load_async_to_lds_b128(
                (AS1 v4i_vs*)gb, (AS3 v4i_vs*)lb, 0, 0);
#else
            *(float4*)la = *(const float4*)ga;
            *(float4*)lb = *(const float4*)gb;
#endif
        }
    };

    v8f acc = {0.f,0.f,0.f,0.f,0.f,0.f,0.f,0.f};

    issue(0, 0);
    for (int kb = 0; kb < nkb; ++kb) {
        int buf = kb & 1;
        bool more = (kb + 1 < nkb);
        if (more) issue(buf ^ 1, kb + 1);     // prefetch next k-block
#if HAVE_ASYNC
        if (more) async_wait<4>();            // in-order: current block done
        else      async_wait<0>();
#endif
        __syncthreads();                      // all waves' staging visible

        const float* as = As[buf];
        const float* bs = Bs[buf];
#pragma unroll
        for (int kk = 0; kk < BKK; kk += 4) {
            v2f a, b;
            int ar = (wm * 16 + lr) * LDSP;
            a.x = as[ar + kk + 2*half];
            a.y = as[ar + kk + 2*half + 1];
            b.x = bs[(kk + 2*half)     * LDSP + wn * 16 + lr];
            b.y = bs[(kk + 2*half + 1) * LDSP + wn * 16 + lr];
            acc = __builtin_amdgcn_wmma_f32_16x16x4_f32(
                      false, a, false, b, (short)0, acc, false, false);
        }
        __syncthreads();                      // done reading buf before reuse
    }

    int n = tn0 + wn * 16 + lr;
    float bv = bias ? bias[n] : 0.0f;
#pragma unroll
    for (int r = 0; r < 8; ++r) {
        int m = tm0 + wm * 16 + half * 8 + r;
        float v = acc[r] + bv;
        if (relu) v = fmaxf(v, 0.0f);
        C[(size_t)m * ldC + n] = v;
    }
}

// =====================================================================
// Attention logits: S[b,h,m,n] = scale * sum_d Q[b,h,m,d] K[b,h,n,d]
// =====================================================================
__global__ void attn_scores_wmma(const float* __restrict__ qkv,
                                 float* __restrict__ S, int Nl, float scale)
{
    int tiles = Nl >> 4;
    int wave = threadIdx.x >> 5;
    int idx = blockIdx.x * (blockDim.x >> 5) + wave;
    int total = BB * HH * tiles * tiles;
    if (idx >= total) return;
    int tn = idx % tiles; idx /= tiles;
    int tm = idx % tiles; idx /= tiles;
    int h  = idx % HH;    int b = idx / HH;

    int lane = threadIdx.x & 31;
    int half = lane >> 4, lr = lane & 15;

    const float* Qrow = qkv + (size_t)(b * Nl + tm * 16 + lr) * (3 * DD) + h * HD;
    const float* Krow = qkv + (size_t)(b * Nl + tn * 16 + lr) * (3 * DD) + DD + h * HD;

    v8f acc = {0.f,0.f,0.f,0.f,0.f,0.f,0.f,0.f};
    for (int k = 0; k < HD; k += 4) {
        v2f a, bfr;
        a.x   = Qrow[k + 2*half];     a.y   = Qrow[k + 2*half + 1];
        bfr.x = Krow[k + 2*half];     bfr.y = Krow[k + 2*half + 1];  // B = K^T
        acc = __builtin_amdgcn_wmma_f32_16x16x4_f32(
                  false, a, false, bfr, (short)0, acc, false, false);
    }
    float* Sb = S + ((size_t)(b * HH + h) * Nl) * Nl;
    int n = tn * 16 + lr;
#pragma unroll
    for (int r = 0; r < 8; ++r) {
        int m = tm * 16 + half * 8 + r;
        Sb[(size_t)m * Nl + n] = acc[r] * scale;
    }
}

// =====================================================================
// Row softmax, one wave per row (wave32 shuffle reductions).
// =====================================================================
__global__ void softmax_rows(float* __restrict__ S, int rows, int Nl)
{
    int wave = threadIdx.x >> 5;
    int row = blockIdx.x * (blockDim.x >> 5) + wave;
    if (row >= rows) return;
    int lane = threadIdx.x & 31;
    float* p = S + (size_t)row * Nl;

    float mx = -3.0e38f;
    for (int j = lane; j < Nl; j += 32) mx = fmaxf(mx, p[j]);
    for (int o = 16; o; o >>= 1) mx = fmaxf(mx, __shfl_xor(mx, o, 32));

    float sum = 0.f;
    for (int j = lane; j < Nl; j += 32) { float e = __expf(p[j] - mx); p[j] = e; sum += e; }
    for (int o = 16; o; o >>= 1) sum += __shfl_xor(sum, o, 32);

    float inv = 1.0f / sum;
    for (int j = lane; j < Nl; j += 32) p[j] *= inv;
}

// =====================================================================
// O = P @ V, written head-concatenated: out[b, m, h*HD + d]
// =====================================================================
__global__ void attn_pv_wmma(const float* __restrict__ qkv,
                             const float* __restrict__ S,
                             float* __restrict__ out, int Nl)
{
    int tilesM = Nl >> 4, tilesN = HD >> 4;
    int wave = threadIdx.x >> 5;
    int idx = blockIdx.x * (blockDim.x >> 5) + wave;
    int total = BB * HH * tilesM * tilesN;
    if (idx >= total) return;
    int tn = idx % tilesN; idx /= tilesN;
    int tm = idx % tilesM; idx /= tilesM;
    int h  = idx % HH;     int b = idx / HH;

    int lane = threadIdx.x & 31;
    int half = lane >> 4, lr = lane & 15;

    const float* Prow  = S + ((size_t)(b * HH + h) * Nl + tm * 16 + lr) * Nl;
    const float* Vbase = qkv + (size_t)(b * Nl) * (3 * DD) + 2 * DD + h * HD + tn * 16 + lr;

    v8f acc = {0.f,0.f,0.f,0.f,0.f,0.f,0.f,0.f};
    for (int k = 0; k < Nl; k += 4) {
        v2f a, bfr;
        a.x   = Prow[k + 2*half];   a.y = Prow[k + 2*half + 1];
        bfr.x = Vbase[(size_t)(k + 2*half)     * (3 * DD)];
        bfr.y = Vbase[(size_t)(k + 2*half + 1) * (3 * DD)];
        acc = __builtin_amdgcn_wmma_f32_16x16x4_f32(
                  false, a, false, bfr, (short)0, acc, false, false);
    }
    int n = tn * 16 + lr;
#pragma unroll
    for (int r = 0; r < 8; ++r) {
        int m = tm * 16 + half * 8 + r;
        out[(size_t)(b * Nl + m) * DD + h * HD + n] = acc[r];
    }
}

// =====================================================================
// pairwise pool: out[r,d] = 0.5*(in[2r,d] + in[2r+1,d])
// =====================================================================
__global__ void pool_pairs(const float* __restrict__ in, float* __restrict__ out,
                           long count)
{
    long i = (long)blockIdx.x * blockDim.x + threadIdx.x;
    if (i >= count) return;
    long r = i / DD; int d = (int)(i % DD);
    out[i] = 0.5f * (in[(2*r) * DD + d] + in[(2*r + 1) * DD + d]);
}

// =====================================================================
// upsample levels 1,2 (repeat_interleave) into concat [B,N,3D] cols D..3D
// =====================================================================
__global__ void upsample_concat(const float* __restrict__ a1,
                                const float* __restrict__ a2,
                                float* __restrict__ concat)
{
    long i = (long)blockIdx.x * blockDim.x + threadIdx.x;
    long total = (long)BB * NN * DD;
    if (i >= total) return;
    int d = (int)(i % DD); long t = i / DD;
    int n = (int)(t % NN); int b = (int)(t / NN);
    size_t crow = ((size_t)b * NN + n) * (3 * DD);
    concat[crow +     DD + d] = a1[((size_t)b * (NN/2) + (n >> 1)) * DD + d];
    concat[crow + 2 * DD + d] = a2[((size_t)b * (NN/4) + (n >> 2)) * DD + d];
}

// =====================================================================
// LayerNorm per row of D, one wave per row
// =====================================================================
__global__ void layernorm_rows(const float* __restrict__ X,
                               const float* __restrict__ g,
                               const float* __restrict__ bb,
                               float* __restrict__ Y, int rows)
{
    int wave = threadIdx.x >> 5;
    int row = blockIdx.x * (blockDim.x >> 5) + wave;
    if (row >= rows) return;
    int lane = threadIdx.x & 31;
    const float* xr = X + (size_t)row * DD;

    float s = 0.f;
    for (int d = lane; d < DD; d += 32) s += xr[d];
    for (int o = 16; o; o >>= 1) s += __shfl_xor(s, o, 32);
    float mean = s * (1.0f / DD);

    float v = 0.f;
    for (int d = lane; d < DD; d += 32) { float t = xr[d] - mean; v += t * t; }
    for (int o = 16; o; o >>= 1) v += __shfl_xor(v, o, 32);
    float inv = rsqrtf(v * (1.0f / DD) + EPS_LN);

    float* yr = Y + (size_t)row * DD;
    for (int d = lane; d < DD; d += 32) yr[d] = (xr[d] - mean) * inv * g[d] + bb[d];
}

// =====================================================================
// edge scores: s[b,i,j] = sigmoid( relu(hi[b,i,:]+hj[b,j,:]) . We2 + be2 )
// =====================================================================
__global__ void edge_scores(const float* __restrict__ hi,
                            const float* __restrict__ hj,
                            const float* __restrict__ We2,
                            const float* __restrict__ be2,
                            float* __restrict__ scores)
{
    int i = blockIdx.x, b = blockIdx.y;
    __shared__ float hs[DD];
    __shared__ float w2[DD];
    for (int d = threadIdx.x; d < DD; d += blockDim.x) {
        hs[d] = hi[((size_t)b * NN + i) * DD + d];
        w2[d] = We2[d];
    }
    __syncthreads();

    int j = threadIdx.x;   // blockDim.x == NN
    const float* hjr = hj + ((size_t)b * NN + j) * DD;
    float acc = 0.f;
    for (int d = 0; d < DD; d += 4) {
        if (d + 64 < DD) __builtin_prefetch(hjr + d + 64, 0, 3);  // global_prefetch_b8
        float4 hv = *(const float4*)(hjr + d);
        acc += fmaxf(hs[d+0] + hv.x, 0.f) * w2[d+0];
        acc += fmaxf(hs[d+1] + hv.y, 0.f) * w2[d+1];
        acc += fmaxf(hs[d+2] + hv.z, 0.f) * w2[d+2];
        acc += fmaxf(hs[d+3] + hv.w, 0.f) * w2[d+3];
    }
    float s = 1.0f / (1.0f + __expf(-(acc + be2[0])));
    if (j == i) s = -1.0f;                 // never selected by top-k (ref sets 0)
    scores[((size_t)b * NN + i) * NN + j] = s;
}

// =====================================================================
// per-row top-K=25 (ties -> lowest index) scattering 1.0 into adjacency
// =====================================================================
__global__ void topk_scatter(const float* __restrict__ scores,
                             float* __restrict__ adj, int rows)
{
    int wave = threadIdx.x >> 5;
    int row = blockIdx.x * (blockDim.x >> 5) + wave;
    if (row >= rows) return;
    int lane = threadIdx.x & 31;

    const float* p = scores + (size_t)row * NN;
    float v[8];
#pragma unroll
    for (int s = 0; s < 8; ++s) v[s] = p[lane + 32 * s];

    float* arow = adj + (size_t)row * NN;
    for (int t = 0; t < KSP; ++t) {
        float bv = -3.0e38f; int bi = 0x7fffffff;
#pragma unroll
        for (int s = 0; s < 8; ++s) {
            int id = lane + 32 * s;
            if (v[s] > bv || (v[s] == bv && id < bi)) { bv = v[s]; bi = id; }
        }
        for (int o = 16; o; o >>= 1) {
            float ov = __shfl_xor(bv, o, 32);
            int   oi = __shfl_xor(bi, o, 32);
            if (ov > bv || (ov == bv && oi < bi)) { bv = ov; bi = oi; }
        }
        if (lane == (bi & 31)) {
            v[bi >> 5] = -3.0e38f;
            arow[bi] = 1.0f;
        }
    }
}

__global__ void zero_kernel(float* __restrict__ p, long n)
{
    long i = (long)blockIdx.x * blockDim.x + threadIdx.x;
    if (i < n) p[i] = 0.f;
}

// =====================================================================
// host-side orchestration
// =====================================================================
static inline int cdiv(long a, long b) { return (int)((a + b - 1) / b); }

extern "C" void kernel_launch(void* const* d_in, const int* in_sizes, int n_in,
                              void* d_out, int out_size, void* d_ws, size_t ws_size,
                              hipStream_t stream)
{
    const float* x      = (const float*)d_in[0];
    // d_in[1] = adjacency (unused by reference)
    const float* Wqkv   = (const float*)d_in[2];
    const float* bqkv   = (const float*)d_in[3];
    const float* Wo     = (const float*)d_in[4];
    const float* bo     = (const float*)d_in[5];
    const float* Wp1    = (const float*)d_in[6];
    const float* bp1    = (const float*)d_in[7];
    const float* Wp2    = (const float*)d_in[8];
    const float* bp2    = (const float*)d_in[9];
    const float* Wfuse  = (const float*)d_in[10];
    const float* bfuse  = (const float*)d_in[11];
    const float* ln_g   = (const float*)d_in[12];
    const float* ln_b   = (const float*)d_in[13];
    const float* We1    = (const float*)d_in[14];
    const float* be1    = (const float*)d_in[15];
    const float* We2    = (const float*)d_in[16];
    const float* be2    = (const float*)d_in[17];
    const float* Ws_qkv = (const float*)d_in[18];
    const float* bs_qkv = (const float*)d_in[19];
    const float* Ws_o   = (const float*)d_in[20];
    const float* bs_o   = (const float*)d_in[21];

    float* attended = (float*)d_out;                       // [B,N,D]
    float* adj_out  = (float*)d_out + (size_t)BB*NN*DD;    // [B,N,N]

    float* f = (float*)d_ws;
    size_t o = 0;
    float* qkv    = f + o; o += (size_t)BB*NN*3*DD;
    float* S      = f + o; o += (size_t)BB*HH*NN*NN;
    float* attn   = f + o; o += (size_t)BB*NN*DD;     // head-concat / hier_pre
    float* cur    = f + o; o += (size_t)BB*NN*DD;
    float* p1     = f + o; o += (size_t)BB*NN*(DD/2);
    float* p2     = f + o; o += (size_t)BB*NN*DD;
    float* concat = f + o; o += (size_t)BB*NN*3*DD;
    float* a1     = f + o; o += (size_t)BB*(NN/2)*DD;
    float* a2     = f + o; o += (size_t)BB*(NN/4)*DD;
    float* hier   = f + o; o += (size_t)BB*NN*DD;
    float* hi     = f + o; o += (size_t)BB*NN*DD;
    float* hj     = f + o; o += (size_t)BB*NN*DD;
    float* scr    = f + o; o += (size_t)BB*NN*NN;

    const float scale = 0.125f;   // HD^-0.5
    const int TPB = 128;

#define GEMM(A, ldA, W, Ncols, K, bias, C, ldC, rowsM, relu)                      \
    do {                                                                          \
        int t32N = (Ncols) / 32;                                                  \
        gemm32_wmma<<<((rowsM) / 32) * t32N, 128, 0, stream>>>(                   \
            (A), (ldA), (W), (Ncols), (K), (bias), (C), (ldC), t32N, (relu));     \
    } while (0)

    // ---------------- hierarchical attention levels ----------------
    const float* curA = x;
    int Nl = NN;
    for (int l = 0; l < LL; ++l) {
        int rows = BB * Nl;
        GEMM(curA, DD, Wqkv + (size_t)l*DD*3*DD, 3*DD, DD, bqkv + (size_t)l*3*DD,
             qkv, 3*DD, rows, 0);

        int tiles = Nl / 16;
        long sc_waves = (long)BB * HH * tiles * tiles;
        attn_scores_wmma<<<cdiv(sc_waves, 4), TPB, 0, stream>>>(qkv, S, Nl, scale);

        long sm_rows = (long)BB * HH * Nl;
        softmax_rows<<<cdiv(sm_rows, 4), TPB, 0, stream>>>(S, (int)sm_rows, Nl);

        long pv_waves = (long)BB * HH * tiles * (HD / 16);
        attn_pv_wmma<<<cdiv(pv_waves, 4), TPB, 0, stream>>>(qkv, S, attn, Nl);

        float* dst; int ldc;
        if (l == 0)      { dst = concat; ldc = 3 * DD; }
        else if (l == 1) { dst = a1;     ldc = DD; }
        else             { dst = a2;     ldc = DD; }
        GEMM(attn, DD, Wo + (size_t)l*DD*DD, DD, DD, bo + (size_t)l*DD,
             dst, ldc, rows, 0);

        if (l < LL - 1) {
            GEMM(dst, ldc, Wp1 + (size_t)l*DD*(DD/2), DD/2, DD, bp1 + (size_t)l*(DD/2),
                 p1, DD/2, rows, 1);
            GEMM(p1, DD/2, Wp2 + (size_t)l*(DD/2)*DD, DD, DD/2, bp2 + (size_t)l*DD,
                 p2, DD, rows, 0);
            long cnt = (long)(rows / 2) * DD;
            pool_pairs<<<cdiv(cnt, 256), 256, 0, stream>>>(p2, cur, cnt);
            curA = cur;
            Nl >>= 1;
        }
    }

    // ---------------- fuse + LayerNorm ----------------
    {
        long cnt = (long)BB * NN * DD;
        upsample_concat<<<cdiv(cnt, 256), 256, 0, stream>>>(a1, a2, concat);
        GEMM(concat, 3*DD, Wfuse, DD, 3*DD, bfuse, attn, DD, BB*NN, 1);   // relu
        layernorm_rows<<<cdiv((long)BB*NN, 4), TPB, 0, stream>>>(attn, ln_g, ln_b,
                                                                 hier, BB*NN);
    }

    // ---------------- sparse edge scoring + top-k ----------------
    GEMM(hier, DD, We1,                 DD, DD, nullptr, hi, DD, BB*NN, 0);
    GEMM(hier, DD, We1 + (size_t)DD*DD, DD, DD, be1,     hj, DD, BB*NN, 0);
    edge_scores<<<dim3(NN, BB), NN, 0, stream>>>(hi, hj, We2, be2, scr);

    long adj_n = (long)BB * NN * NN;
    zero_kernel<<<cdiv(adj_n, 256), 256, 0, stream>>>(adj_out, adj_n);
    topk_scatter<<<cdiv((long)BB*NN, 4), TPB, 0, stream>>>(scr, adj_out, BB*NN);

    // ---------------- final MHA on hier -> attended ----------------
    GEMM(hier, DD, Ws_qkv, 3*DD, DD, bs_qkv, qkv, 3*DD, BB*NN, 0);
    {
        int tiles = NN / 16;
        long sc_waves = (long)BB * HH * tiles * tiles;
        attn_scores_wmma<<<cdiv(sc_waves, 4), TPB, 0, stream>>>(qkv, S, NN, scale);
        long sm_rows = (long)BB * HH * NN;
        softmax_rows<<<cdiv(sm_rows, 4), TPB, 0, stream>>>(S, (int)sm_rows, NN);
        long pv_waves = (long)BB * HH * tiles * (HD / 16);
        attn_pv_wmma<<<cdiv(pv_waves, 4), TPB, 0, stream>>>(qkv, S, attn, NN);
    }
    GEMM(attn, DD, Ws_o, DD, DD, bs_o, attended, DD, BB*NN, 0);

#undef GEMM
    (void)in_sizes; (void)n_in; (void)out_size; (void)ws_size;
}